// ME_Voxelization_40973988004386
// MI455X (gfx1250) — compile-verified
//
#include <hip/hip_runtime.h>
#include <hip/hip_bf16.h>
#include <stdint.h>

// ---------------- problem constants ----------------
#define N_POINTS   4194304            // 4M points
#define NBLK       1024               // sort blocks
#define BLK_ITEMS  4096               // items per sort block (NBLK*BLK_ITEMS == N)
#define THREADS    256                // 8 waves (wave32)
#define TILES      (BLK_ITEMS / THREADS)
#define RADIX      256
#define HIST_ENTRIES (RADIX * NBLK)   // 262144
#define SCAN_CHUNKS  (HIST_ENTRIES / 1024)  // 256

typedef unsigned int  uint32x4 __attribute__((ext_vector_type(4)));
typedef int           int32x8  __attribute__((ext_vector_type(8)));
typedef int           int32x4  __attribute__((ext_vector_type(4)));

__device__ __forceinline__ uint32_t lane_ballot32(bool p) {
#if __has_builtin(__builtin_amdgcn_ballot_w32)
  return __builtin_amdgcn_ballot_w32(p);
#else
  return (uint32_t)__ballot(p);
#endif
}

// 30-bit packed voxel key: lexicographic in (x,y,z), same order as the
// reference's 60-bit key, so sorted-unique output is identical.
__device__ __forceinline__ int voxel_coord(float v) {
  int c = (int)floorf(v / 0.1f);
  c = c < -512 ? -512 : c;
  c = c >  511 ?  511 : c;
  return c;
}

// ---------------- stage 1: TDM-staged key generation ----------------
__global__ __launch_bounds__(THREADS)
void voxel_keys_kernel(const float* __restrict__ pts, uint32_t* __restrict__ keys) {
  __shared__ float lds[BLK_ITEMS * 3];       // 48 KB staging tile
  const uint32_t b = blockIdx.x;
  const uint32_t t = threadIdx.x;
  const size_t baseElem = (size_t)b * (BLK_ITEMS * 3);

#if defined(__gfx1250__) && __has_builtin(__builtin_amdgcn_tensor_load_to_lds) && __has_builtin(__builtin_amdgcn_s_wait_tensorcnt)
  if ((t >> 5) == 0) {  // wave 0 only (uniform branch) issues the TDM DMA
    uint64_t gaddr   = (uint64_t)(uintptr_t)(pts + baseElem);
    uint32_t ldsAddr = (uint32_t)(uintptr_t)&lds[0];
    const uint32_t TD0 = (uint32_t)N_POINTS * 3u;     // tensor_dim0 (elements)
    uint32x4 g0;
    g0[0] = 1u;                                        // count=1 user descriptor
    g0[1] = ldsAddr;                                   // lds_addr
    g0[2] = (uint32_t)(gaddr & 0xFFFFFFFFu);           // global_addr[31:0]
    g0[3] = (uint32_t)((gaddr >> 32) & 0x1FFFFFFu)     // global_addr[56:32]
          | (2u << 30);                                // type=2 ("image")
    int32x8 g1;
    g1[0] = (int)(2u << 16);                           // data_size=4B; wg_mask=0
    g1[1] = (int)((TD0 & 0xFFFFu) << 16);              // tensor_dim0[15:0]
    g1[2] = (int)(((TD0 >> 16) & 0xFFFFu) | (1u << 16)); // td0[31:16] | tensor_dim1=1
    g1[3] = (int)((uint32_t)(BLK_ITEMS * 3) << 16);    // tile_dim0 = 12288
    g1[4] = 1;                                         // tile_dim1=1, tile_dim2=0
    g1[5] = (int)TD0;                                  // tensor_dim0_stride[31:0]
    g1[6] = 0; g1[7] = 0;
    int32x4 z4 = {0, 0, 0, 0};
    int32x8 z8 = {0, 0, 0, 0, 0, 0, 0, 0};
    // clang-23 / therock-10.0 6-arg form: (g0, g1, g2, g3, extra_group, cpol)
    __builtin_amdgcn_tensor_load_to_lds(g0, g1, z4, z4, z8, 0);
    __builtin_amdgcn_s_wait_tensorcnt(0);
  }
  __syncthreads();
#else
  for (int i = t; i < BLK_ITEMS * 3; i += THREADS) lds[i] = pts[baseElem + i];
  __syncthreads();
#endif

  for (int i = 0; i < TILES; ++i) {
    int e = i * THREADS + (int)t;
    int xi = voxel_coord(lds[3 * e + 0]);
    int yi = voxel_coord(lds[3 * e + 1]);
    int zi = voxel_coord(lds[3 * e + 2]);
    uint32_t key = ((uint32_t)(xi + 512) << 20)
                 | ((uint32_t)(yi + 512) << 10)
                 |  (uint32_t)(zi + 512);
    keys[(size_t)b * BLK_ITEMS + e] = key;
  }
}

// ---------------- stage 2: stable LSD radix sort (4 x 8-bit passes) ----------
__global__ __launch_bounds__(THREADS)
void radix_hist_kernel(const uint32_t* __restrict__ keys, uint32_t* __restrict__ hist,
                       int shift) {
  __shared__ uint32_t lh[RADIX];
  const uint32_t t = threadIdx.x, b = blockIdx.x;
  lh[t] = 0;
  __syncthreads();
  const size_t base = (size_t)b * BLK_ITEMS;
  for (int i = 0; i < TILES; ++i) {
    uint32_t k = keys[base + i * THREADS + t];
    atomicAdd(&lh[(k >> shift) & 255u], 1u);
  }
  __syncthreads();
  hist[(size_t)t * NBLK + b] = lh[t];   // digit-major layout for global scan
}

// --- hierarchical exclusive scan (3 kernels), replaces single-WGP sweep ---
// level 1: each 1024-thread block exclusively scans its 1024-entry chunk in
// place and emits the chunk total.
__global__ __launch_bounds__(1024)
void scan_chunk_kernel(uint32_t* __restrict__ data, uint32_t* __restrict__ sums) {
  __shared__ uint32_t s[1024];
  const uint32_t t = threadIdx.x;
  const uint32_t idx = blockIdx.x * 1024u + t;
  uint32_t v = data[idx];
  s[t] = v;
  __syncthreads();
  for (int off = 1; off < 1024; off <<= 1) {
    uint32_t tmp = (t >= (uint32_t)off) ? s[t - off] : 0u;
    __syncthreads();
    s[t] += tmp;
    __syncthreads();
  }
  data[idx] = s[t] - v;                       // exclusive within chunk
  if (t == 1023) sums[blockIdx.x] = s[1023];  // chunk total
}

// level 2: single-block exclusive scan of n (<=1024) totals, in place.
__global__ __launch_bounds__(1024)
void scan_single_kernel(uint32_t* __restrict__ data, int n) {
  __shared__ uint32_t s[1024];
  const uint32_t t = threadIdx.x;
  uint32_t v = data[t];
  s[t] = v;
  __syncthreads();
  for (int off = 1; off < n; off <<= 1) {
    uint32_t tmp = (t >= (uint32_t)off) ? s[t - off] : 0u;
    __syncthreads();
    s[t] += tmp;
    __syncthreads();
  }
  data[t] = s[t] - v;                         // exclusive
}

// level 3: add scanned chunk offsets back.
__global__ __launch_bounds__(1024)
void add_offsets_kernel(uint32_t* __restrict__ data, const uint32_t* __restrict__ sums) {
  data[blockIdx.x * 1024u + threadIdx.x] += sums[blockIdx.x];
}

__global__ __launch_bounds__(THREADS)
void radix_scatter_kernel(const uint32_t* __restrict__ in, uint32_t* __restrict__ out,
                          const uint32_t* __restrict__ hist, int shift) {
  __shared__ uint32_t localBase[RADIX];
  __shared__ uint32_t waveCnt[8][RADIX];
  const uint32_t t = threadIdx.x, b = blockIdx.x;
  localBase[t] = hist[(size_t)t * NBLK + b];
  const uint32_t wave = t >> 5, lane = t & 31;
  const uint32_t ltmask = (lane == 0) ? 0u : (0xFFFFFFFFu >> (32 - lane));
  const size_t base = (size_t)b * BLK_ITEMS;

  for (int tile = 0; tile < TILES; ++tile) {
    for (int w = 0; w < 8; ++w) waveCnt[w][t] = 0;
    __syncthreads();
    uint32_t k = in[base + tile * THREADS + t];
    uint32_t d = (k >> shift) & 255u;
    // wave32 digit match via ballot (stable within wave by lane order)
    uint32_t mask = 0xFFFFFFFFu;
    for (int bit = 0; bit < 8; ++bit) {
      uint32_t bal = lane_ballot32(((d >> bit) & 1u) != 0u);
      mask &= ((d >> bit) & 1u) ? bal : ~bal;
    }
    uint32_t waveRank = __popc(mask & ltmask);
    if (waveRank == 0) waveCnt[wave][d] = __popc(mask);  // one leader per digit
    __syncthreads();
    uint32_t prefix = 0;
    for (uint32_t w = 0; w < wave; ++w) prefix += waveCnt[w][d];
    out[localBase[d] + prefix + waveRank] = k;
    __syncthreads();
    uint32_t tot = 0;
    for (int w = 0; w < 8; ++w) tot += waveCnt[w][t];
    localBase[t] += tot;
    __syncthreads();
  }
}

// ---------------- stage 3: unique-compact + decode ----------------
__global__ void zero_out_kernel(int* __restrict__ coords, unsigned char* __restrict__ valid) {
  size_t i = (size_t)blockIdx.x * blockDim.x + threadIdx.x;
  if (i < (size_t)N_POINTS) {
    coords[3 * i + 0] = 0;
    coords[3 * i + 1] = 0;
    coords[3 * i + 2] = 0;
    valid[i] = 0;
  }
}

__global__ __launch_bounds__(THREADS)
void flag_scan_kernel(const uint32_t* __restrict__ sorted, uint32_t* __restrict__ ep,
                      uint32_t* __restrict__ blockSums) {
  __shared__ uint32_t s[THREADS];
  const uint32_t t = threadIdx.x, b = blockIdx.x;
  const size_t start = (size_t)b * BLK_ITEMS + (size_t)t * TILES;
  uint32_t kk[TILES];
  for (int i = 0; i < TILES; ++i) kk[i] = sorted[start + i];
  uint32_t prev = (start == 0) ? ~kk[0] : sorted[start - 1];
  uint32_t cnt = 0, ff = 0;
  for (int i = 0; i < TILES; ++i) {
    uint32_t f = (kk[i] != prev) ? 1u : 0u;
    ff |= f << i;
    cnt += f;
    prev = kk[i];
  }
  s[t] = cnt;
  __syncthreads();
  for (int off = 1; off < THREADS; off <<= 1) {
    uint32_t tmp = (t >= (uint32_t)off) ? s[t - off] : 0u;
    __syncthreads();
    s[t] += tmp;
    __syncthreads();
  }
  uint32_t run = s[t] - cnt;                 // thread-exclusive prefix
  if (t == THREADS - 1) blockSums[b] = s[THREADS - 1];
  for (int i = 0; i < TILES; ++i) {
    ep[start + i] = run;
    run += (ff >> i) & 1u;
  }
}

__global__ void unique_scatter_kernel(const uint32_t* __restrict__ sorted,
                                      const uint32_t* __restrict__ ep,
                                      const uint32_t* __restrict__ bs,
                                      int* __restrict__ coords,
                                      unsigned char* __restrict__ valid) {
  size_t i = (size_t)blockIdx.x * blockDim.x + threadIdx.x;
  if (i >= (size_t)N_POINTS) return;
  uint32_t k = sorted[i];
  uint32_t prev = (i == 0) ? ~k : sorted[i - 1];
  if (k != prev) {
    size_t pos = (size_t)bs[i >> 12] + ep[i];   // 4096 items per flag block
    coords[3 * pos + 0] = (int)(k >> 20) - 512;
    coords[3 * pos + 1] = (int)((k >> 10) & 1023u) - 512;
    coords[3 * pos + 2] = (int)(k & 1023u) - 512;
    valid[pos] = 1;
  }
}

// ---------------- launch ----------------
extern "C" void kernel_launch(void* const* d_in, const int* in_sizes, int n_in,
                              void* d_out, int out_size, void* d_ws, size_t ws_size,
                              hipStream_t stream) {
  const float* pts = (const float*)d_in[0];

  // workspace layout (u32):
  //   keysA[4M] | keysB[4M] | hist[262144] | bsums[1024] | hsums[256]
  uint32_t* keysA = (uint32_t*)d_ws;
  uint32_t* keysB = keysA + N_POINTS;
  uint32_t* hist  = keysB + N_POINTS;
  uint32_t* bsums = hist + HIST_ENTRIES;
  uint32_t* hsums = bsums + 1024;

  int* coords = (int*)d_out;                                      // [N,3] int32
  unsigned char* valid = (unsigned char*)(coords + (size_t)3 * N_POINTS); // [N] bool

  voxel_keys_kernel<<<NBLK, THREADS, 0, stream>>>(pts, keysA);

  uint32_t* bufs[2] = {keysA, keysB};
  int cur = 0;
  for (int pass = 0; pass < 4; ++pass) {
    int shift = pass * 8;
    radix_hist_kernel<<<NBLK, THREADS, 0, stream>>>(bufs[cur], hist, shift);
    // hierarchical exclusive scan of the 262144-entry digit-major histogram
    scan_chunk_kernel<<<SCAN_CHUNKS, 1024, 0, stream>>>(hist, hsums);
    scan_single_kernel<<<1, SCAN_CHUNKS, 0, stream>>>(hsums, SCAN_CHUNKS);
    add_offsets_kernel<<<SCAN_CHUNKS, 1024, 0, stream>>>(hist, hsums);
    radix_scatter_kernel<<<NBLK, THREADS, 0, stream>>>(bufs[cur], bufs[cur ^ 1], hist, shift);
    cur ^= 1;
  }
  uint32_t* sorted = bufs[cur];       // even #passes -> back in keysA
  uint32_t* ep     = bufs[cur ^ 1];   // reuse other buffer for per-item prefixes

  zero_out_kernel<<<(N_POINTS + 255) / 256, 256, 0, stream>>>(coords, valid);
  flag_scan_kernel<<<NBLK, THREADS, 0, stream>>>(sorted, ep, bsums);
  scan_single_kernel<<<1, 1024, 0, stream>>>(bsums, 1024);
  unique_scatter_kernel<<<(N_POINTS + 255) / 256, 256, 0, stream>>>(sorted, ep, bsums,
                                                                    coords, valid);
}